// ResolutionDecoder_70153995813113
// MI455X (gfx1250) — compile-verified
//
#include <hip/hip_runtime.h>
#include <hip/hip_bf16.h>
#include <math.h>
#include <stdint.h>

#ifndef __has_builtin
#define __has_builtin(x) 0
#endif

// ---------------------------------------------------------------------------
// Shapes (fixed by the reference setup)
// ---------------------------------------------------------------------------
static constexpr int V   = 32000;
static constexpr int Th  = 1024;
static constexpr int Tu  = 256;
static constexpr int B   = 32;
static constexpr int H   = 512;
static constexpr int E   = 256;
static constexpr int VT  = V + Th;      // 33024 (proba / u_copy row length)
static constexpr int M2  = 2 * V + Th;  // 65024 (joint softmax length)
static constexpr int PAD = 4;           // LDS row padding (floats) -> bank spread

typedef float v2f __attribute__((ext_vector_type(2)));
typedef float v8f __attribute__((ext_vector_type(8)));
typedef int   v4i __attribute__((ext_vector_type(4)));

// address-space qualified int4 pointers for the async global->LDS DMA builtin
typedef __attribute__((address_space(1))) v4i as1_v4i;
typedef __attribute__((address_space(3))) v4i as3_v4i;

// D = A(16x4) * B(4x16) + C(16x16), fp32 WMMA (wave32)
__device__ __forceinline__ v8f wmma_f32(v2f a, v2f b, v8f c) {
  return __builtin_amdgcn_wmma_f32_16x16x4_f32(
      /*neg_a=*/false, a, /*neg_b=*/false, b,
      /*c_mod=*/(short)0, c, /*reuse_a=*/false, /*reuse_b=*/false);
}

// 16B global -> LDS copy. Prefer the CDNA5 async-DMA path (ASYNCcnt);
// fall back to a register round-trip if the builtin is absent.
__device__ __forceinline__ void async_b128(const float* g, float* l) {
#if __has_builtin(__builtin_amdgcn_global_load_async_to_lds_b128)
  __builtin_amdgcn_global_load_async_to_lds_b128(
      (as1_v4i*)(unsigned long long)(uintptr_t)g,   // global address (AS1)
      (as3_v4i*)(unsigned int)(uintptr_t)l,         // LDS offset = addr[31:0] (AS3)
      0, 0);
#else
  float4 v = *(const float4*)g;
  *(float4*)l = v;
#endif
}

__device__ __forceinline__ void wait_async_lds() {
#if __has_builtin(__builtin_amdgcn_global_load_async_to_lds_b128)
#if __has_builtin(__builtin_amdgcn_s_wait_asynccnt)
  __builtin_amdgcn_s_wait_asynccnt(0);
#else
  asm volatile("s_wait_asynccnt 0x0" ::: "memory");
#endif
#endif
  __syncthreads();   // also covers the fallback ds_store path
}

// ---------------------------------------------------------------------------
// Kernel 1: attention logits
//   energy = tanh([h0, enc] @ attn_W^T + attn_b); logit[b,t] = dot(attn_v, energy)
//   One wave32 per (b, 16-row t-tile). K = 2H = 1024, N = H = 512.
//   A tile staged in LDS once (h0 shared across rows), read 32x from LDS.
//   K-loop split at the h0/enc boundary -> fixed LDS base per loop, no
//   per-iteration address cndmask in the WMMA inner loop.
// ---------------------------------------------------------------------------
__global__ void attn_logits_kernel(const float* __restrict__ u_enc_out, // (Tu,B,H)
                                   const float* __restrict__ h0,        // (B,H)
                                   const float* __restrict__ attn_W,    // (H,2H)
                                   const float* __restrict__ attn_b,    // (H)
                                   const float* __restrict__ attn_v,    // (H)
                                   float* __restrict__ logits)          // (B,Tu)
{
  const int tt   = blockIdx.x;       // Tu/16 tiles
  const int b    = blockIdx.y;
  const int lane = threadIdx.x & 31;
  const int half = lane >> 4;        // 0: K even-pair, 1: K odd-pair
  const int mr   = lane & 15;
  const int t0   = tt * 16;

  __shared__ __attribute__((aligned(16))) float h0s[H];            // k < H (all rows)
  __shared__ __attribute__((aligned(16))) float encs[16][H + PAD]; // k >= H, per row

  // ---- stage A tile into LDS (async DMA) ----
  {
    const float* h0b = h0 + (size_t)b * H;
    for (int c = lane; c < H / 4; c += 32)
      async_b128(h0b + c * 4, &h0s[c * 4]);
    for (int c = lane; c < 16 * (H / 4); c += 32) {
      const int row = c >> 7;            // c / (H/4)
      const int col = (c & 127) << 2;    // (c % (H/4)) * 4
      async_b128(u_enc_out + ((size_t)(t0 + row) * B + b) * H + col,
                 &encs[row][col]);
    }
    wait_async_lds();
  }

  const float* aRow = &encs[mr][0];   // per-lane enc row base (fixed)

  float acc[8];
#pragma unroll
  for (int r = 0; r < 8; ++r) acc[r] = 0.f;

  for (int nt = 0; nt < H / 16; ++nt) {
    const int n = nt * 16 + mr;                 // C column this lane owns
    const float* Wrow = attn_W + (size_t)n * (2 * H);
    v8f c = {};
    // K in [0, H): A comes from the shared h0 vector
    for (int k0 = 0; k0 < H; k0 += 4) {
      const int ka = k0 + half * 2;
      v2f a, bb;
      a.x = h0s[ka];     a.y = h0s[ka + 1];
      bb.x = Wrow[ka];   bb.y = Wrow[ka + 1];
      c = wmma_f32(a, bb, c);
    }
    // K in [H, 2H): A comes from this t-row's enc slice
    for (int k0 = 0; k0 < H; k0 += 4) {
      const int ka = k0 + half * 2;
      v2f a, bb;
      a.x = aRow[ka];         a.y = aRow[ka + 1];
      bb.x = Wrow[H + ka];    bb.y = Wrow[H + ka + 1];
      c = wmma_f32(a, bb, c);
    }
    const float bias = attn_b[n];
    const float vsc  = attn_v[n];
#pragma unroll
    for (int r = 0; r < 8; ++r) acc[r] += tanhf(c[r] + bias) * vsc;
  }

  // reduce the 16 columns held by each half-wave
#pragma unroll
  for (int r = 0; r < 8; ++r) {
    float v = acc[r];
#pragma unroll
    for (int off = 1; off < 16; off <<= 1) v += __shfl_xor(v, off, 32);
    acc[r] = v;
  }
  if (mr == 0) {
#pragma unroll
    for (int r = 0; r < 8; ++r)
      logits[b * Tu + t0 + r + 8 * half] = acc[r];
  }
}

// ---------------------------------------------------------------------------
// Kernel 2: attention softmax + context  (one block per b, 256 threads = Tu)
// ---------------------------------------------------------------------------
__global__ void att_context_kernel(const float* __restrict__ logits,
                                   const float* __restrict__ u_enc_out,
                                   float* __restrict__ u_context) // (B,H)
{
  const int b   = blockIdx.x;
  const int tid = threadIdx.x;     // 256
  __shared__ float satt[Tu];
  __shared__ float red[256];

  float l = logits[b * Tu + tid];
  red[tid] = l; __syncthreads();
  for (int s = 128; s > 0; s >>= 1) { if (tid < s) red[tid] = fmaxf(red[tid], red[tid + s]); __syncthreads(); }
  const float mx = red[0]; __syncthreads();
  const float e = expf(l - mx);
  red[tid] = e; __syncthreads();
  for (int s = 128; s > 0; s >>= 1) { if (tid < s) red[tid] += red[tid + s]; __syncthreads(); }
  satt[tid] = e / red[0];
  __syncthreads();

  for (int h = tid; h < H; h += 256) {
    float acc = 0.f;
    for (int t = 0; t < Tu; ++t)
      acc += satt[t] * u_enc_out[((size_t)t * B + b) * H + h];
    u_context[b * H + h] = acc;
  }
}

// ---------------------------------------------------------------------------
// Kernel 3: GRU cell + p_gen  (one block per b, H=512 threads)
// ---------------------------------------------------------------------------
__global__ void gru_kernel(const int* __restrict__ c_t_input,   // (1,B)
                           const float* __restrict__ emb,       // (V,E)
                           const float* __restrict__ u_context, // (B,H)
                           const float* __restrict__ h0,        // (B,H)
                           const float* __restrict__ W_ih,      // (3H, E+H)
                           const float* __restrict__ W_hh,      // (3H, H)
                           const float* __restrict__ b_ih,
                           const float* __restrict__ b_hh,
                           const float* __restrict__ pgen_W,    // (1, 2H+E)
                           const float* __restrict__ pgen_b,    // (1)
                           float* __restrict__ h_new_ws,        // (B,H)
                           float* __restrict__ out_newhidden,   // (B,H)
                           float* __restrict__ out_gruout,      // (B,H)
                           float* __restrict__ p1_ws)           // (B)
{
  const int b = blockIdx.x;
  const int h = threadIdx.x;   // 512
  const int tok = c_t_input[b];
  const float* er  = emb + (size_t)tok * E;
  const float* uc  = u_context + (size_t)b * H;
  const float* h0b = h0 + (size_t)b * H;

  float gr = b_ih[h], gz = b_ih[H + h], gn = b_ih[2 * H + h];
  const int KX = E + H; // 768
  const float* wr = W_ih + (size_t)h * KX;
  const float* wz = W_ih + (size_t)(H + h) * KX;
  const float* wn = W_ih + (size_t)(2 * H + h) * KX;
  for (int k = 0; k < E; ++k) {
    const float xv = er[k];
    gr += wr[k] * xv;  gz += wz[k] * xv;  gn += wn[k] * xv;
  }
  for (int k = 0; k < H; ++k) {
    const float xv = uc[k];
    gr += wr[E + k] * xv;  gz += wz[E + k] * xv;  gn += wn[E + k] * xv;
  }
  float hr = b_hh[h], hz = b_hh[H + h], hn = b_hh[2 * H + h];
  for (int k = 0; k < H; ++k) {
    const float hv = h0b[k];
    hr += W_hh[(size_t)h * H + k] * hv;
    hz += W_hh[(size_t)(H + h) * H + k] * hv;
    hn += W_hh[(size_t)(2 * H + h) * H + k] * hv;
  }
  const float r = 1.f / (1.f + expf(-(gr + hr)));
  const float z = 1.f / (1.f + expf(-(gz + hz)));
  const float n = tanhf(gn + r * hn);
  const float hnew = (1.f - z) * n + z * h0b[h];
  h_new_ws[b * H + h]      = hnew;
  out_newhidden[b * H + h] = hnew;
  out_gruout[b * H + h]    = hnew;

  // p_gen = sigmoid([u_context, h_new, c_embed] . pgen_W + pgen_b)
  __shared__ float red[H];
  float part = uc[h] * pgen_W[h] + hnew * pgen_W[H + h];
  if (h < E) part += er[h] * pgen_W[2 * H + h];
  red[h] = part; __syncthreads();
  for (int s = H / 2; s > 0; s >>= 1) { if (h < s) red[h] += red[h + s]; __syncthreads(); }
  if (h == 0) p1_ws[b] = 1.f / (1.f + expf(-(red[0] + pgen_b[0])));
}

// ---------------------------------------------------------------------------
// Kernel 4: gen_score = [u_context, h_new] @ proj_W^T + proj_b
//   WMMA f32. One wave32 per (16-col vocab tile, 16-row b tile). K = 1024.
//   proj_W (131 MB, single use) is streamed non-temporally to keep the
//   reused tensors (hist 64MB + u_enc 16MB) resident in the 192MB L2.
//   K-loop split at the u_context/h_new boundary (static bases).
// ---------------------------------------------------------------------------
__global__ void gen_score_kernel(const float* __restrict__ u_context, // (B,H)
                                 const float* __restrict__ h_new,     // (B,H)
                                 const float* __restrict__ proj_W,    // (V,2H)
                                 const float* __restrict__ proj_b,    // (V)
                                 float* __restrict__ gen)             // (B,V)
{
  const int ntile = blockIdx.x;       // V/16 = 2000
  const int mtile = blockIdx.y;       // 0..1  (B=32)
  const int lane  = threadIdx.x & 31;
  const int half  = lane >> 4;
  const int mr    = lane & 15;
  const int n     = ntile * 16 + mr;
  const int brow  = mtile * 16 + mr;  // A row = batch index

  const float* Wrow = proj_W + (size_t)n * (2 * H);
  const float* ucR  = u_context + (size_t)brow * H;
  const float* hnR  = h_new + (size_t)brow * H;

  v8f c = {};
  for (int k0 = 0; k0 < H; k0 += 4) {
    __builtin_prefetch(Wrow + k0 + 256, 0, 0);
    const int ka = k0 + half * 2;
    v2f a, bb;
    a.x = ucR[ka];  a.y = ucR[ka + 1];
    bb.x = __builtin_nontemporal_load(Wrow + ka);
    bb.y = __builtin_nontemporal_load(Wrow + ka + 1);
    c = wmma_f32(a, bb, c);
  }
  for (int k0 = 0; k0 < H; k0 += 4) {
    __builtin_prefetch(Wrow + H + k0 + 256, 0, 0);
    const int ka = k0 + half * 2;
    v2f a, bb;
    a.x = hnR[ka];  a.y = hnR[ka + 1];
    bb.x = __builtin_nontemporal_load(Wrow + H + ka);
    bb.y = __builtin_nontemporal_load(Wrow + H + ka + 1);
    c = wmma_f32(a, bb, c);
  }
  const float bias = proj_b[n];
#pragma unroll
  for (int r = 0; r < 8; ++r) {
    const int b = mtile * 16 + r + 8 * half;
    gen[(size_t)b * V + n] = c[r] + bias;
  }
}

// ---------------------------------------------------------------------------
// Kernel 5: copy scores
//   cs[b,t] = dot(tanh(hist[t,b,:] @ copy2_W^T + copy2_b), h_new[b,:])
//   WMMA f32, fused tanh + dot epilogue. A tile staged once in LDS via
//   async DMA (kills the 32x global re-read of history_enc_out).
// ---------------------------------------------------------------------------
__global__ void copy_score_kernel(const float* __restrict__ hist,    // (Th,B,H)
                                  const float* __restrict__ copy2_W, // (H,H)
                                  const float* __restrict__ copy2_b, // (H)
                                  const float* __restrict__ h_new,   // (B,H)
                                  float* __restrict__ cs)            // (B,Th)
{
  const int tt   = blockIdx.x;       // Th/16 = 64
  const int b    = blockIdx.y;
  const int lane = threadIdx.x & 31;
  const int half = lane >> 4;
  const int mr   = lane & 15;
  const int t0   = tt * 16;

  __shared__ __attribute__((aligned(16))) float as[16][H + PAD];

  // ---- stage A tile (16 history rows) into LDS ----
  for (int c = lane; c < 16 * (H / 4); c += 32) {
    const int row = c >> 7;            // c / (H/4)
    const int col = (c & 127) << 2;    // (c % (H/4)) * 4
    async_b128(hist + ((size_t)(t0 + row) * B + b) * H + col, &as[row][col]);
  }
  wait_async_lds();

  const float* aRow = &as[mr][0];     // per-lane history row base (fixed)

  float acc[8];
#pragma unroll
  for (int r = 0; r < 8; ++r) acc[r] = 0.f;

  for (int nt = 0; nt < H / 16; ++nt) {
    const int n = nt * 16 + mr;
    const float* Wrow = copy2_W + (size_t)n * H;
    v8f c = {};
    for (int k0 = 0; k0 < H; k0 += 4) {
      const int ka = k0 + half * 2;
      v2f a, bb;
      a.x = aRow[ka];    a.y = aRow[ka + 1];
      bb.x = Wrow[ka];   bb.y = Wrow[ka + 1];
      c = wmma_f32(a, bb, c);
    }
    const float bias = copy2_b[n];
    const float hv   = h_new[b * H + n];
#pragma unroll
    for (int r = 0; r < 8; ++r) acc[r] += tanhf(c[r] + bias) * hv;
  }

#pragma unroll
  for (int r = 0; r < 8; ++r) {
    float v = acc[r];
#pragma unroll
    for (int off = 1; off < 16; off <<= 1) v += __shfl_xor(v, off, 32);
    acc[r] = v;
  }
  if (mr == 0) {
#pragma unroll
    for (int r = 0; r < 8; ++r)
      cs[b * Th + t0 + r + 8 * half] = acc[r];
  }
}

// ---------------------------------------------------------------------------
// Kernel 6: copy distribution row (exp + scatter-add), one block per b
// ---------------------------------------------------------------------------
__global__ void copy_dist_kernel(const float* __restrict__ cs,  // (B,Th)
                                 const int* __restrict__ hdn,   // (Th,B)
                                 float* __restrict__ e_ws,      // (B,Th)
                                 float* __restrict__ cmax_ws,   // (B)
                                 float* __restrict__ outrow)    // (B, V+Th)
{
  const int b   = blockIdx.x;
  const int tid = threadIdx.x;   // 256
  __shared__ float red[256];

  float mx = -INFINITY;
  for (int t = tid; t < Th; t += 256) mx = fmaxf(mx, cs[b * Th + t]);
  red[tid] = mx; __syncthreads();
  for (int s = 128; s > 0; s >>= 1) { if (tid < s) red[tid] = fmaxf(red[tid], red[tid + s]); __syncthreads(); }
  mx = red[0]; __syncthreads();

  float sum = 0.f;
  for (int t = tid; t < Th; t += 256) {
    const float ev = expf(cs[b * Th + t] - mx);
    e_ws[b * Th + t] = ev;
    sum += ev;
  }
  red[tid] = sum; __syncthreads();
  for (int s = 128; s > 0; s >>= 1) { if (tid < s) red[tid] += red[tid + s]; __syncthreads(); }
  const float base = 1e-10f * red[0];
  if (tid == 0) cmax_ws[b] = mx;
  __syncthreads();

  float* row = outrow + (size_t)b * VT;
  for (int j = tid; j < VT; j += 256) row[j] = base;
  __syncthreads();

  for (int t = tid; t < Th - 1; t += 256) {
    const int w = hdn[t * B + b];
    const bool req = (w >= 10 && w <= 14);                  // REQ_IDS
    const bool oov = (w == 2) || (w >= V);
    const int col = req ? (w + 10) : (oov ? (V + t) : w);   // SLOT_OFFSET=10
    const float val = (oov && !req) ? 5.0f : 1.0f;
    atomicAdd(&row[col], e_ws[b * Th + t + 1] * (val - 1e-10f));
  }
}

// ---------------------------------------------------------------------------
// Kernel 7: joint softmax over [gen*p1 | u_copy*p2] and recombination
// ---------------------------------------------------------------------------
__global__ void final_softmax_kernel(const float* __restrict__ gen,     // (B,V)
                                     const float* __restrict__ outrow,  // (B,V+Th)
                                     const float* __restrict__ cmax_ws, // (B)
                                     const float* __restrict__ p1_ws,   // (B)
                                     float* __restrict__ proba)         // (B,V+Th)
{
  const int b   = blockIdx.x;
  const int tid = threadIdx.x;  // 512
  const float p1  = p1_ws[b];
  const float p2  = 1.f - p1;
  const float cmx = cmax_ws[b];
  const float* g   = gen + (size_t)b * V;
  const float* row = outrow + (size_t)b * VT;
  __shared__ float red[512];

  float mx = -INFINITY;
  for (int j = tid; j < M2; j += 512) {
    const float l = (j < V) ? g[j] * p1 : (logf(row[j - V]) + cmx) * p2;
    mx = fmaxf(mx, l);
  }
  red[tid] = mx; __syncthreads();
  for (int s = 256; s > 0; s >>= 1) { if (tid < s) red[tid] = fmaxf(red[tid], red[tid + s]); __syncthreads(); }
  mx = red[0]; __syncthreads();

  float sum = 0.f;
  for (int j = tid; j < M2; j += 512) {
    const float l = (j < V) ? g[j] * p1 : (logf(row[j - V]) + cmx) * p2;
    sum += expf(l - mx);
  }
  red[tid] = sum; __syncthreads();
  for (int s = 256; s > 0; s >>= 1) { if (tid < s) red[tid] += red[tid + s]; __syncthreads(); }
  const float inv = 1.f / red[0];

  float* pb = proba + (size_t)b * VT;
  for (int j = tid; j < VT; j += 512) {
    const float lc = (logf(row[j]) + cmx) * p2;
    const float c  = expf(lc - mx) * inv;
    pb[j] = (j < V) ? (expf(g[j] * p1 - mx) * inv + c) : c;
  }
}

// ---------------------------------------------------------------------------
// Host launcher
// ---------------------------------------------------------------------------
extern "C" void kernel_launch(void* const* d_in, const int* in_sizes, int n_in,
                              void* d_out, int out_size, void* d_ws, size_t ws_size,
                              hipStream_t stream) {
  (void)in_sizes; (void)n_in; (void)out_size; (void)ws_size;

  const int*   hdn      = (const int*)  d_in[0];   // (Th,B)
  const float* hist     = (const float*)d_in[1];   // (Th,B,H)
  const float* u_enc    = (const float*)d_in[2];   // (Tu,B,H)
  /* d_in[3] = u_input_np, unused by the reference */
  const int*   c_t      = (const int*)  d_in[4];   // (1,B)
  const float* lasth    = (const float*)d_in[5];   // (1,B,H) -> h0
  const float* emb      = (const float*)d_in[6];   // (V,E)
  const float* attn_W   = (const float*)d_in[7];
  const float* attn_b   = (const float*)d_in[8];
  const float* attn_v   = (const float*)d_in[9];
  const float* W_ih     = (const float*)d_in[10];
  const float* W_hh     = (const float*)d_in[11];
  const float* b_ih     = (const float*)d_in[12];
  const float* b_hh     = (const float*)d_in[13];
  const float* proj_W   = (const float*)d_in[14];
  const float* proj_b   = (const float*)d_in[15];
  const float* copy2_W  = (const float*)d_in[16];
  const float* copy2_b  = (const float*)d_in[17];
  const float* pgen_W   = (const float*)d_in[18];
  const float* pgen_b   = (const float*)d_in[19];

  float* out = (float*)d_out;
  float* proba        = out;                         // B*VT
  float* out_newhid   = out + (size_t)B * VT;        // B*H
  float* out_gruout   = out_newhid + (size_t)B * H;  // B*H

  // workspace layout (floats)
  float* ws = (float*)d_ws;
  float* ws_logits = ws;                       size_t o = (size_t)B * Tu;
  float* ws_uctx   = ws + o;                   o += (size_t)B * H;
  float* ws_hnew   = ws + o;                   o += (size_t)B * H;
  float* ws_p1     = ws + o;                   o += B;
  float* ws_cmax   = ws + o;                   o += B;
  float* ws_gen    = ws + o;                   o += (size_t)B * V;
  float* ws_cs     = ws + o;                   o += (size_t)B * Th;
  float* ws_e      = ws + o;                   o += (size_t)B * Th;
  float* ws_outrow = ws + o;                   o += (size_t)B * VT;

  // 1) attention logits (WMMA f32, LDS-staged A via async DMA)
  attn_logits_kernel<<<dim3(Tu / 16, B), dim3(32), 0, stream>>>(
      u_enc, lasth, attn_W, attn_b, attn_v, ws_logits);

  // 2) attention softmax + context
  att_context_kernel<<<dim3(B), dim3(256), 0, stream>>>(ws_logits, u_enc, ws_uctx);

  // 3) GRU + p_gen
  gru_kernel<<<dim3(B), dim3(H), 0, stream>>>(
      c_t, emb, ws_uctx, lasth, W_ih, W_hh, b_ih, b_hh, pgen_W, pgen_b,
      ws_hnew, out_newhid, out_gruout, ws_p1);

  // 4) vocab projection GEMM (WMMA f32, non-temporal 131 MB stream)
  gen_score_kernel<<<dim3(V / 16, B / 16), dim3(32), 0, stream>>>(
      ws_uctx, ws_hnew, proj_W, proj_b, ws_gen);

  // 5) copy-score GEMM (WMMA f32, LDS-staged A via async DMA, fused tanh+dot)
  copy_score_kernel<<<dim3(Th / 16, B), dim3(32), 0, stream>>>(
      hist, copy2_W, copy2_b, ws_hnew, ws_cs);

  // 6) copy distribution row
  copy_dist_kernel<<<dim3(B), dim3(256), 0, stream>>>(
      ws_cs, hdn, ws_e, ws_cmax, ws_outrow);

  // 7) joint softmax + recombine
  final_softmax_kernel<<<dim3(B), dim3(512), 0, stream>>>(
      ws_gen, ws_outrow, ws_cmax, ws_p1, proba);
}